// LSTMModel_76020921140418
// MI455X (gfx1250) — compile-verified
//
#include <hip/hip_runtime.h>
#include <hip/hip_bf16.h>
#include <math.h>

typedef __attribute__((ext_vector_type(2))) float v2f;
typedef __attribute__((ext_vector_type(8))) float v8f;

#define HID   100
#define BATCH 16
#define NGATE 400
#define TLEN  256
#define MROWS (TLEN * BATCH)   // 4096
#define VOCAB_N 30522

// ---------------------------------------------------------------------------
// Kernel 1: gx[m, n] = sum_k emb[input[m], k] * dec_W_ih[n, k] + b_ih[n] + b_hh[n]
// M = 4096, N = 400, K = 100. One wave per 16x16 tile. grid = (25, 256).
// ---------------------------------------------------------------------------
__global__ __launch_bounds__(32) void gx_gemm_kernel(
    const int* __restrict__ tok,      // [M] token id per row (embedding gather)
    const float* __restrict__ emb,    // [VOCAB, 100]
    const float* __restrict__ W,      // [400, 100]
    const float* __restrict__ b0,     // [400]
    const float* __restrict__ b1,     // [400]
    float* __restrict__ out)          // [M, 400]
{
    const int lane = threadIdx.x;
    const int tn   = blockIdx.x;          // N tile (0..24)
    const int tm   = blockIdx.y;          // M tile (0..255)
    const int half = lane >> 4;           // 0 | 1
    const int l16  = lane & 15;
    const int koff = half * 2;            // A/B k sub-offset per ISA layout
    const int n    = tn * 16 + l16;       // B column this lane feeds

    const int   mA   = tm * 16 + l16;     // A row this lane feeds
    const float* aRow = emb + (size_t)tok[mA] * HID;
    const float* wRow = W   + (size_t)n * HID;

    const float bias = b0[n] + b1[n];
    v8f acc;
#pragma unroll
    for (int j = 0; j < 8; ++j) acc[j] = bias;

#pragma unroll
    for (int s = 0; s < 25; ++s) {
        const int kb = s * 4 + koff;
        v2f a = *(const v2f*)(aRow + kb);
        v2f b = *(const v2f*)(wRow + kb);
        acc = __builtin_amdgcn_wmma_f32_16x16x4_f32(
            false, a, false, b, (short)0, acc, false, false);
    }

    float* o = out + (size_t)(tm * 16 + half * 8) * NGATE + tn * 16 + l16;
#pragma unroll
    for (int j = 0; j < 8; ++j) o[(size_t)j * NGATE] = acc[j];
}

// ---------------------------------------------------------------------------
// Kernel 2: LSTM recurrence. Single workgroup, 800 threads = 25 waves.
// Wave w owns gate columns [w*16, w*16+16). W_hh slice register-resident.
// h, c, gates in LDS. 256 serial steps, WMMA f32 16x16x4 for h @ W_hh^T.
// ---------------------------------------------------------------------------
__global__ __launch_bounds__(800) void lstm_kernel(
    const float* __restrict__ gx,      // [T*16, 400] precomputed input gates
    const float* __restrict__ Whh,     // [400, 100]
    float* __restrict__ dec_out)       // [T*16, 100]
{
    __shared__ float h_lds[BATCH][HID];
    __shared__ float c_lds[BATCH][HID];
    __shared__ float g_lds[BATCH][NGATE];

    const int tid  = threadIdx.x;
    const int lane = tid & 31;
    const int wave = tid >> 5;          // 0..24 -> N tile
    const int half = lane >> 4;
    const int l16  = lane & 15;
    const int koff = half * 2;
    const int n    = wave * 16 + l16;   // gate column 0..399

    // Preload this wave's W_hh tile into registers: 25 x float2 = 50 VGPRs.
    v2f wreg[25];
    {
        const float* wRow = Whh + (size_t)n * HID;
#pragma unroll
        for (int s = 0; s < 25; ++s)
            wreg[s] = *(const v2f*)(wRow + s * 4 + koff);
    }

    // h0 = c0 = 0
    for (int p = tid; p < BATCH * HID; p += 800) {
        ((float*)h_lds)[p] = 0.0f;
        ((float*)c_lds)[p] = 0.0f;
    }
    __syncthreads();

    for (int t = 0; t < TLEN; ++t) {
        // C init from precomputed gx (includes both biases).
        const float* gxt = gx + (size_t)t * BATCH * NGATE;
        v8f acc;
#pragma unroll
        for (int j = 0; j < 8; ++j)
            acc[j] = gxt[(size_t)(j + half * 8) * NGATE + n];

        // acc += h @ W_hh^T for this 16-column gate tile.
#pragma unroll
        for (int s = 0; s < 25; ++s) {
            v2f a;
            a.x = h_lds[l16][s * 4 + koff];
            a.y = h_lds[l16][s * 4 + koff + 1];
            acc = __builtin_amdgcn_wmma_f32_16x16x4_f32(
                false, a, false, wreg[s], (short)0, acc, false, false);
        }

        // Spill gates tile to LDS so i/f/g/o (different waves) can be combined.
#pragma unroll
        for (int j = 0; j < 8; ++j)
            g_lds[j + half * 8][n] = acc[j];
        __syncthreads();

        // Elementwise gate math: 16*100 = 1600 cells over 800 threads.
#pragma unroll
        for (int r = 0; r < 2; ++r) {
            const int p   = tid * 2 + r;        // 0..1599
            const int m   = p / HID;
            const int hid = p % HID;
            const float gi = g_lds[m][hid];
            const float gf = g_lds[m][HID + hid];
            const float gg = g_lds[m][2 * HID + hid];
            const float go = g_lds[m][3 * HID + hid];
            const float si = 1.0f / (1.0f + __expf(-gi));
            const float sf = 1.0f / (1.0f + __expf(-gf));
            const float so = 1.0f / (1.0f + __expf(-go));
            const float c  = sf * c_lds[m][hid] + si * tanhf(gg);
            const float h  = so * tanhf(c);
            c_lds[m][hid] = c;
            h_lds[m][hid] = h;
            dec_out[((size_t)t * BATCH + m) * HID + hid] = h;
        }
        __syncthreads();   // order h writes vs. next step's WMMA reads
    }
}

// ---------------------------------------------------------------------------
// Kernel 3: out[m, v] = sum_k dec_out[m, k] * out_W[v, k] + out_b[v]
// M = 4096, N = 30522, K = 100. 4 waves/block, 1 N-tile per wave.
// grid = (477, 256); 477*4 = 1908 = ceil(30522/16). Store-bound (~500 MB out).
// ---------------------------------------------------------------------------
__global__ __launch_bounds__(128) void out_gemm_kernel(
    const float* __restrict__ A,       // [M, 100]
    const float* __restrict__ W,       // [VOCAB, 100]
    const float* __restrict__ bias,    // [VOCAB]
    float* __restrict__ out,           // [M, VOCAB]
    int N)
{
    const int lane = threadIdx.x & 31;
    const int wave = threadIdx.x >> 5;          // 0..3
    const int tn   = blockIdx.x * 4 + wave;     // N tile (0..1907)
    const int tm   = blockIdx.y;                // M tile (0..255)
    const int half = lane >> 4;
    const int l16  = lane & 15;
    const int koff = half * 2;

    const int n  = tn * 16 + l16;
    const int nc = (n < N) ? n : (N - 1);       // clamp loads; keep EXEC full for WMMA

    const float* aRow = A + (size_t)(tm * 16 + l16) * HID;
    const float* wRow = W + (size_t)nc * HID;

    const float bv = bias[nc];
    v8f acc;
#pragma unroll
    for (int j = 0; j < 8; ++j) acc[j] = bv;

#pragma unroll
    for (int s = 0; s < 25; ++s) {
        const int kb = s * 4 + koff;
        v2f a = *(const v2f*)(aRow + kb);
        v2f b = *(const v2f*)(wRow + kb);
        acc = __builtin_amdgcn_wmma_f32_16x16x4_f32(
            false, a, false, b, (short)0, acc, false, false);
    }

    if (n < N) {
        float* o = out + (size_t)(tm * 16 + half * 8) * N + n;
#pragma unroll
        for (int j = 0; j < 8; ++j) o[(size_t)j * N] = acc[j];
    }
}

// ---------------------------------------------------------------------------
// Launch: gx GEMM -> recurrence -> vocab projection (stream-ordered).
// Encoder LSTM from the reference is dead code (outputs unused) and skipped.
// ---------------------------------------------------------------------------
extern "C" void kernel_launch(void* const* d_in, const int* in_sizes, int n_in,
                              void* d_out, int out_size, void* d_ws, size_t ws_size,
                              hipStream_t stream) {
    (void)in_sizes; (void)n_in; (void)out_size; (void)ws_size;

    const int*   input    = (const int*)  d_in[0];
    const float* emb      = (const float*)d_in[1];
    const float* dec_W_ih = (const float*)d_in[6];
    const float* dec_W_hh = (const float*)d_in[7];
    const float* dec_b_ih = (const float*)d_in[8];
    const float* dec_b_hh = (const float*)d_in[9];
    const float* out_W    = (const float*)d_in[10];
    const float* out_b    = (const float*)d_in[11];
    float*       out      = (float*)d_out;

    float* gx      = (float*)d_ws;                 // 4096*400 f32 = 6.4 MB
    float* dec_out = gx + (size_t)MROWS * NGATE;   // 4096*100 f32 = 1.6 MB

    gx_gemm_kernel<<<dim3(NGATE / 16, MROWS / 16), 32, 0, stream>>>(
        input, emb, dec_W_ih, dec_b_ih, dec_b_hh, gx);

    lstm_kernel<<<1, 800, 0, stream>>>(gx, dec_W_hh, dec_out);

    out_gemm_kernel<<<dim3(477, MROWS / 16), 128, 0, stream>>>(
        dec_out, out_W, out_b, out, VOCAB_N);
}